// HyperGNN_81157702025498
// MI455X (gfx1250) — compile-verified
//
#include <hip/hip_runtime.h>

// HyperGNN: 3x (scatter-sum over edges -> GEMM(W)+bias+ReLU) -> GEMM(W_out)+bias
// N=100000 nodes, E=600000 edges, D=D_OUT=128.
// Feature matrix (51.2MB) is L2-resident (192MB L2) -> atomic scatter resolves
// in L2. GEMM uses fp32 WMMA (V_WMMA_F32_16X16X4_F32) for reference fidelity:
// 64 rows/block, 4 accumulator tiles per wave sharing each B fragment (4x W
// reuse + 4 independent WMMA chains). A tile staged to LDS via
// GLOBAL_LOAD_ASYNC_TO_LDS_B128 when the builtin is available.

#define FD 128  // feature dim (D == D_OUT == 128)

#if defined(__has_builtin)
#if __has_builtin(__builtin_amdgcn_global_load_async_to_lds_b128)
#define HG_ASYNC_LDS 1
#endif
#endif

typedef __attribute__((ext_vector_type(2))) float v2f;
typedef __attribute__((ext_vector_type(8))) float v8f;
typedef __attribute__((__vector_size__(4 * sizeof(int)))) int hg_v4i;

__global__ void hg_zero_kernel(float4* __restrict__ p, long long n4) {
  long long i = (long long)blockIdx.x * blockDim.x + threadIdx.x;
  long long stride = (long long)gridDim.x * blockDim.x;
  float4 z; z.x = 0.f; z.y = 0.f; z.z = 0.f; z.w = 0.f;
  for (; i < n4; i += stride) p[i] = z;
}

// One 32-lane group per edge: lane handles 4 consecutive features.
__global__ void hg_scatter_add_kernel(const float* __restrict__ x,
                                      const long long* __restrict__ src,
                                      const long long* __restrict__ dst,
                                      float* __restrict__ agg, int E) {
  long long t = (long long)blockIdx.x * blockDim.x + threadIdx.x;
  int e = (int)(t >> 5);
  if (e >= E) return;
  int lane = (int)(t & 31);
  int s = (int)src[e];
  int d = (int)dst[e];
  const float4 v = *(const float4*)(x + (size_t)s * FD + lane * 4);
  float* o = agg + (size_t)d * FD + lane * 4;
  atomicAdd(o + 0, v.x);
  atomicAdd(o + 1, v.y);
  atomicAdd(o + 2, v.z);
  atomicAdd(o + 3, v.w);
}

// out[r0..r0+63, :] = act(A[r0..r0+63, :] @ Wm + bias)
// 8 waves/block; wave w owns output columns [16w,16w+16) for all 4 row tiles.
// A tile staged fully in LDS before any store -> safe for out == A (in-place).
__global__ __launch_bounds__(256) void hg_gemm_bias_act_kernel(
    const float* __restrict__ A, const float* __restrict__ Wm,
    const float* __restrict__ bias, float* __restrict__ out,
    int N, int do_relu) {
  __shared__ __align__(16) float As[64 * FD];  // 32KB

  const int tid = threadIdx.x;
  const int r0 = blockIdx.x * 64;

  // Stage 64x128 fp32 A tile: 2048 float4, 8 per thread (coalesced).
  // Zero-pad rows past N.
#pragma unroll
  for (int j = 0; j < 8; ++j) {
    int q = j * 256 + tid;
    int row = q >> 5;  // 32 float4 per row
    int c4 = q & 31;
    int gr = r0 + row;
    if (gr < N) {
#ifdef HG_ASYNC_LDS
      __builtin_amdgcn_global_load_async_to_lds_b128(
          (hg_v4i __attribute__((address_space(1)))*)(A + (size_t)gr * FD + c4 * 4),
          (hg_v4i __attribute__((address_space(3)))*)(&As[row * FD + c4 * 4]),
          /*offset=*/0, /*cpol=*/0);
#else
      *(float4*)(&As[row * FD + c4 * 4]) =
          *(const float4*)(A + (size_t)gr * FD + c4 * 4);
#endif
    } else {
      float4 z; z.x = 0.f; z.y = 0.f; z.z = 0.f; z.w = 0.f;
      *(float4*)(&As[row * FD + c4 * 4]) = z;
    }
  }
#ifdef HG_ASYNC_LDS
  asm volatile("s_wait_asynccnt 0" ::: "memory");  // barrier doesn't cover ASYNCcnt
#endif
  __syncthreads();

  const int w = tid >> 5;
  const int lane = tid & 31;
  const int hi = lane >> 4;       // lane half: K offset (A/B) / row+8 (C)
  const int m = lane & 15;        // A row within tile / B N-col
  const int col = (w << 4) + m;   // this lane's global output column
  const int koff = hi << 1;       // K offset within the 4-wide K slab

  // gfx1250 prefetch path for the (hot, L2-resident) weight matrix
  __builtin_prefetch(Wm + (size_t)lane * FD, 0, 3);

  v8f acc[4];
#pragma unroll
  for (int t = 0; t < 4; ++t)
    acc[t] = (v8f){0.f, 0.f, 0.f, 0.f, 0.f, 0.f, 0.f, 0.f};

#pragma unroll
  for (int k = 0; k < FD; k += 4) {
    // B 4x16 fp32 fragment: lane (=N col m) holds rows K=k+koff, k+koff+1
    v2f bfrag;
    bfrag.x = Wm[(size_t)(k + koff) * FD + col];
    bfrag.y = Wm[(size_t)(k + koff + 1) * FD + col];
#pragma unroll
    for (int t = 0; t < 4; ++t) {
      // A 16x4 fp32 fragment (ISA 7.12.2): lane m holds K=k+koff, k+koff+1
      v2f a;
      a.x = As[(t * 16 + m) * FD + k + koff];
      a.y = As[(t * 16 + m) * FD + k + koff + 1];
      acc[t] = __builtin_amdgcn_wmma_f32_16x16x4_f32(
          /*neg_a=*/false, a, /*neg_b=*/false, bfrag,
          /*c_mod=*/(short)0, acc[t], /*reuse_a=*/false, /*reuse_b=*/false);
    }
  }

  const float bv = bias[col];
#pragma unroll
  for (int t = 0; t < 4; ++t) {
    const int rbase = r0 + t * 16 + (hi << 3);  // C: VGPR i -> row i / i+8
#pragma unroll
    for (int i = 0; i < 8; ++i) {
      int row = rbase + i;
      float v = acc[t][i] + bv;
      if (do_relu) v = fmaxf(v, 0.f);
      if (row < N) out[(size_t)row * FD + col] = v;
    }
  }
}

extern "C" void kernel_launch(void* const* d_in, const int* in_sizes, int n_in,
                              void* d_out, int out_size, void* d_ws, size_t ws_size,
                              hipStream_t stream) {
  const float* x        = (const float*)d_in[0];
  const long long* ei   = (const long long*)d_in[1];  // int64 edge_index [2,E]
  const float* W        = (const float*)d_in[2];
  const float* b        = (const float*)d_in[3];
  const float* W_out    = (const float*)d_in[4];
  const float* b_out    = (const float*)d_in[5];
  float* out            = (float*)d_out;

  const int N = in_sizes[0] / FD;
  const int E = in_sizes[1] / 2;
  const long long* src = ei;
  const long long* dst = ei + E;

  float* bufA = (float*)d_ws;  // needs N*FD*4 = 51.2MB of workspace
  float* bufB = out;           // reuse output buffer as second ping-pong buffer

  const long long n4 = (long long)N * FD / 4;
  const int zgrid = 4096;
  const int sgrid = (int)(((long long)E * 32 + 255) / 256);
  const int ggrid = (N + 63) / 64;

  // layer 1: agg(x) -> bufA, h1 = relu(bufA@W+b) in-place in bufA
  hg_zero_kernel<<<zgrid, 256, 0, stream>>>((float4*)bufA, n4);
  hg_scatter_add_kernel<<<sgrid, 256, 0, stream>>>(x, src, dst, bufA, E);
  hg_gemm_bias_act_kernel<<<ggrid, 256, 0, stream>>>(bufA, W, b, bufA, N, 1);

  // layer 2: agg(h1) -> bufB, h2 in-place in bufB
  hg_zero_kernel<<<zgrid, 256, 0, stream>>>((float4*)bufB, n4);
  hg_scatter_add_kernel<<<sgrid, 256, 0, stream>>>(bufA, src, dst, bufB, E);
  hg_gemm_bias_act_kernel<<<ggrid, 256, 0, stream>>>(bufB, W, b, bufB, N, 1);

  // layer 3: agg(h2) -> bufA, h3 in-place in bufA
  hg_zero_kernel<<<zgrid, 256, 0, stream>>>((float4*)bufA, n4);
  hg_scatter_add_kernel<<<sgrid, 256, 0, stream>>>(bufB, src, dst, bufA, E);
  hg_gemm_bias_act_kernel<<<ggrid, 256, 0, stream>>>(bufA, W, b, bufA, N, 1);

  // output layer: out = h3 @ W_out + b_out (no relu)
  hg_gemm_bias_act_kernel<<<ggrid, 256, 0, stream>>>(bufA, W_out, b_out, out, N, 0);
}